// HerbMultiInstanceEncoder_54245436948916
// MI455X (gfx1250) — compile-verified
//
#include <hip/hip_runtime.h>
#include <hip/hip_bf16.h>
#include <math.h>

#define DD 128
#define HERBS 25000

typedef float v2f __attribute__((ext_vector_type(2)));
typedef float v8f __attribute__((ext_vector_type(8)));

// ---------------- init ----------------
__global__ void zero_f32_kernel(float* __restrict__ p, long n) {
  long i = (long)blockIdx.x * blockDim.x + threadIdx.x;
  if (i < n) p[i] = 0.f;
}

__global__ void init_seg_kernel(int* cntA, int* startA, int* cntB, int* startB, int H) {
  int i = blockIdx.x * blockDim.x + threadIdx.x;
  if (i < H) { cntA[i] = 0; startA[i] = 0x7fffffff; cntB[i] = 0; startB[i] = 0x7fffffff; }
}

// Pack row-major weight W[K][N] into pair-interleaved layout so a (k,k+1) column
// pair is one contiguous 8B load: dst[((k>>1)*N + n)*2 + (k&1)] = src[k*N + n]
__global__ void pack_pairs_kernel(const float* __restrict__ src, float* __restrict__ dst,
                                  int K, int N) {
  long i = (long)blockIdx.x * blockDim.x + threadIdx.x;
  long total = (long)K * N;
  if (i < total) {
    int k = (int)(i / N), n = (int)(i % N);
    dst[(((long)(k >> 1)) * N + n) * 2 + (k & 1)] = src[i];
  }
}

// ---------------- segment sums / counts / starts (indices sorted) ----------------
__global__ void __launch_bounds__(128) seg_accum_kernel(
    const float* __restrict__ h, const int* __restrict__ idx,
    float* __restrict__ sum, int* __restrict__ cnt, int* __restrict__ start, int N) {
  __shared__ int sidx[64];
  int c0 = blockIdx.x * 64;
  int t = threadIdx.x;                 // 0..127 == feature dim
  int nrows = N - c0; if (nrows > 64) nrows = 64;
  if (nrows <= 0) return;
  if (t < nrows) sidx[t] = idx[c0 + t];
  __syncthreads();
  float acc = 0.f;
  int cur = sidx[0];
  for (int r = 0; r < nrows; ++r) {
    int hh = sidx[r];
    if (hh != cur) { atomicAdd(&sum[(long)cur * DD + t], acc); acc = 0.f; cur = hh; }
    acc += h[((long)(c0 + r)) * DD + t];
  }
  atomicAdd(&sum[(long)cur * DD + t], acc);
  if (t == 0) {
    int rs = 0; cur = sidx[0];
    for (int r = 1; r <= nrows; ++r) {
      int hh = (r < nrows) ? sidx[r] : -1;
      if (hh != cur) {
        atomicAdd(&cnt[cur], r - rs);
        atomicMin(&start[cur], c0 + rs);
        rs = r; cur = hh;
      }
    }
  }
}

__global__ void __launch_bounds__(128) seg_mean_kernel(float* __restrict__ sum,
                                                       const int* __restrict__ cnt) {
  int hh = blockIdx.x; int t = threadIdx.x;
  int c = cnt[hh];
  float inv = 1.f / (float)(c > 1 ? c : 1);
  sum[(long)hh * DD + t] *= inv;
}

// ---------------- attention scores: WMMA f32 16x16x4, 32 rows/wave ----------------
// X[row] = concat(h[row], meanO[idx[row]])  (256)   H1 = X @ aw1 (256x64)
// score = lrelu(H1 + ab1) @ aw2 + ab2
// K loop split in two affine halves: [0,128) from h, [128,256) from meanO.
__global__ void __launch_bounds__(256) attn_score_kernel(
    const float* __restrict__ h, const int* __restrict__ idx,
    const float* __restrict__ meanO,
    const float* __restrict__ aw1, const float* __restrict__ ab1,
    const float* __restrict__ aw2, const float* __restrict__ ab2,
    float* __restrict__ score, int N, int numTiles32) {
  __shared__ float W1s[256 * 64];                   // 64 KB, pair-interleaved
  for (int i = threadIdx.x; i < 256 * 64; i += 256) {
    int k = i >> 6, col = i & 63;
    W1s[((((k >> 1) << 6) + col) << 1) + (k & 1)] = aw1[i];
  }
  __syncthreads();
  int wave = threadIdx.x >> 5;
  int lane = threadIdx.x & 31;
  int t32 = blockIdx.x * 8 + wave;
  if (t32 >= numTiles32) return;
  int half = lane >> 4;                             // K-half select for A/B frags
  int m16  = lane & 15;
  int row0 = t32 * 32 + m16;      if (row0 >= N) row0 = N - 1;   // clamp (loads only)
  int row1 = t32 * 32 + 16 + m16; if (row1 >= N) row1 = N - 1;
  const float* p0 = h + (long)row0 * DD;
  const float* p1 = h + (long)row1 * DD;
  const float* q0 = meanO + (long)idx[row0] * DD;
  const float* q1 = meanO + (long)idx[row1] * DD;
  v8f acc0[4] = {}, acc1[4] = {};
  // first half: X[:,0:128] = h rows
  for (int k0 = 0; k0 < DD; k0 += 4) {
    int ka = k0 + 2 * half;                         // even; ka,ka+1 contiguous
    v2f a0 = *(const v2f*)(p0 + ka);
    v2f a1 = *(const v2f*)(p1 + ka);
    int kp = (ka >> 1) << 6;
#pragma unroll
    for (int nt = 0; nt < 4; ++nt) {
      int col = nt * 16 + m16;
      v2f b = *(const v2f*)&W1s[(kp + col) << 1];
      acc0[nt] = __builtin_amdgcn_wmma_f32_16x16x4_f32(false, a0, false, b,
                                                       (short)0, acc0[nt], false, false);
      acc1[nt] = __builtin_amdgcn_wmma_f32_16x16x4_f32(false, a1, false, b,
                                                       (short)0, acc1[nt], false, false);
    }
  }
  // second half: X[:,128:256] = meanO[idx]
  for (int k0 = 0; k0 < DD; k0 += 4) {
    int ka = k0 + 2 * half;
    v2f a0 = *(const v2f*)(q0 + ka);
    v2f a1 = *(const v2f*)(q1 + ka);
    int kp = (((ka >> 1) + 64) << 6);               // K offset +128 in pair units
#pragma unroll
    for (int nt = 0; nt < 4; ++nt) {
      int col = nt * 16 + m16;
      v2f b = *(const v2f*)&W1s[(kp + col) << 1];
      acc0[nt] = __builtin_amdgcn_wmma_f32_16x16x4_f32(false, a0, false, b,
                                                       (short)0, acc0[nt], false, false);
      acc1[nt] = __builtin_amdgcn_wmma_f32_16x16x4_f32(false, a1, false, b,
                                                       (short)0, acc1[nt], false, false);
    }
  }
  // fused bias + LeakyReLU(0.2) + 64->1 projection, for both 16-row tiles
  float pr0[8], pr1[8];
#pragma unroll
  for (int r = 0; r < 8; ++r) { pr0[r] = 0.f; pr1[r] = 0.f; }
#pragma unroll
  for (int nt = 0; nt < 4; ++nt) {
    int col = nt * 16 + m16;
    float b1 = ab1[col], w2 = aw2[col];
#pragma unroll
    for (int r = 0; r < 8; ++r) {
      float x0 = acc0[nt][r] + b1; x0 = (x0 > 0.f) ? x0 : 0.2f * x0;
      float x1 = acc1[nt][r] + b1; x1 = (x1 > 0.f) ? x1 : 0.2f * x1;
      pr0[r] += x0 * w2;
      pr1[r] += x1 * w2;
    }
  }
#pragma unroll
  for (int m = 1; m < 16; m <<= 1) {
#pragma unroll
    for (int r = 0; r < 8; ++r) {
      pr0[r] += __shfl_xor(pr0[r], m, 32);
      pr1[r] += __shfl_xor(pr1[r], m, 32);
    }
  }
  if (m16 == 0) {
    float b2 = ab2[0];
#pragma unroll
    for (int r = 0; r < 8; ++r) {                   // D layout: VGPR r -> M=r / M=r+8
      int o0 = t32 * 32 + r + 8 * half;
      int o1 = o0 + 16;
      if (o0 < N) score[o0] = pr0[r] + b2;
      if (o1 < N) score[o1] = pr1[r] + b2;
    }
  }
}

// ---------------- per-herb softmax + weighted sum ----------------
__global__ void __launch_bounds__(128) seg_attn_kernel(
    const float* __restrict__ h, const float* __restrict__ score,
    const int* __restrict__ start, const int* __restrict__ cnt,
    float* __restrict__ out) {
  int hh = blockIdx.x;
  int t = threadIdx.x;                              // feature dim
  int c = cnt[hh];
  if (c == 0) { out[(long)hh * DD + t] = 0.f; return; }
  int s0 = start[hh];
  __shared__ float red[128];
  float m = -3.4e38f;
  for (int r = t; r < c; r += 128) m = fmaxf(m, score[s0 + r]);
  red[t] = m; __syncthreads();
  for (int s = 64; s > 0; s >>= 1) { if (t < s) red[t] = fmaxf(red[t], red[t + s]); __syncthreads(); }
  m = red[0]; __syncthreads();
  float dsum = 0.f;
  for (int r = t; r < c; r += 128) dsum += __expf(score[s0 + r] - m);
  red[t] = dsum; __syncthreads();
  for (int s = 64; s > 0; s >>= 1) { if (t < s) red[t] += red[t + s]; __syncthreads(); }
  float denom = red[0] + 1e-16f;
  float acc = 0.f;
  for (int r = 0; r < c; ++r) {
    float wv = __expf(score[s0 + r] - m);
    acc += wv * h[(long)(s0 + r) * DD + t];          // coalesced 512B row per step
  }
  out[(long)hh * DD + t] = acc / denom;
}

// ---------------- fusion MLP (WMMA f32, pair-interleaved weights) ----------------
// 4 WMMAs against one A frag, B from pair-interleaved global weights
__device__ __forceinline__ void wmma4(v2f a, const float* __restrict__ Bp, long kp,
                                      int colBase, int m16, v8f acc[4]) {
#pragma unroll
  for (int nt = 0; nt < 4; ++nt) {
    int col = colBase + nt * 16 + m16;
    v2f b = *(const v2f*)&Bp[(kp + col) * 2];
    acc[nt] = __builtin_amdgcn_wmma_f32_16x16x4_f32(false, a, false, b,
                                                    (short)0, acc[nt], false, false);
  }
}

__global__ void __launch_bounds__(256) fusion1_kernel(
    const float* __restrict__ HA, const float* __restrict__ HB,
    const float* __restrict__ fw1p, const float* __restrict__ fb1,
    float* __restrict__ mid, int H, int nJobs) {
  int wave = threadIdx.x >> 5, lane = threadIdx.x & 31;
  int job = blockIdx.x * 8 + wave;
  if (job >= nJobs) return;
  int mt = job >> 2, ng = job & 3;                   // 4 n-groups of 64 cols
  int half = lane >> 4, m16 = lane & 15;
  int row = mt * 16 + m16; if (row >= H) row = H - 1;
  const float* pA = HA + (long)row * DD;
  const float* pB = HB + (long)row * DD;
  int cb = ng * 64;
  v8f acc[4] = {};
  // section 0: HA
  for (int k0 = 0; k0 < DD; k0 += 4) {
    int ka = k0 + 2 * half;
    wmma4(*(const v2f*)(pA + ka), fw1p, (long)(ka >> 1) * 256, cb, m16, acc);
  }
  // section 1: HB
  for (int k0 = 0; k0 < DD; k0 += 4) {
    int ka = k0 + 2 * half;
    wmma4(*(const v2f*)(pB + ka), fw1p, (long)((ka >> 1) + 64) * 256, cb, m16, acc);
  }
  // section 2: HA * HB
  for (int k0 = 0; k0 < DD; k0 += 4) {
    int ka = k0 + 2 * half;
    v2f a = *(const v2f*)(pA + ka), b2 = *(const v2f*)(pB + ka);
    v2f a2; a2.x = a.x * b2.x; a2.y = a.y * b2.y;
    wmma4(a2, fw1p, (long)((ka >> 1) + 128) * 256, cb, m16, acc);
  }
  // section 3: |HA - HB|
  for (int k0 = 0; k0 < DD; k0 += 4) {
    int ka = k0 + 2 * half;
    v2f a = *(const v2f*)(pA + ka), b2 = *(const v2f*)(pB + ka);
    v2f a2; a2.x = fabsf(a.x - b2.x); a2.y = fabsf(a.y - b2.y);
    wmma4(a2, fw1p, (long)((ka >> 1) + 192) * 256, cb, m16, acc);
  }
#pragma unroll
  for (int nt = 0; nt < 4; ++nt) {
    int col = cb + nt * 16 + m16;
    float bias = fb1[col];
#pragma unroll
    for (int r = 0; r < 8; ++r) {
      int orow = mt * 16 + r + 8 * half;
      if (orow < H) {
        float x = acc[nt][r] + bias;
        mid[(long)orow * 256 + col] = (x > 0.f) ? x : 0.f;   // ReLU
      }
    }
  }
}

__global__ void __launch_bounds__(256) fusion2_kernel(
    const float* __restrict__ mid, const float* __restrict__ fw2p,
    const float* __restrict__ fb2, float* __restrict__ out, int H, int nJobs) {
  int wave = threadIdx.x >> 5, lane = threadIdx.x & 31;
  int job = blockIdx.x * 8 + wave;
  if (job >= nJobs) return;
  int mt = job >> 1, ng = job & 1;                   // 2 n-groups of 64 cols
  int half = lane >> 4, m16 = lane & 15;
  int row = mt * 16 + m16; if (row >= H) row = H - 1;
  const float* pM = mid + (long)row * 256;
  int cb = ng * 64;
  v8f acc[4] = {};
  for (int k0 = 0; k0 < 256; k0 += 4) {
    int ka = k0 + 2 * half;
    wmma4(*(const v2f*)(pM + ka), fw2p, (long)(ka >> 1) * DD, cb, m16, acc);
  }
#pragma unroll
  for (int nt = 0; nt < 4; ++nt) {
    int col = cb + nt * 16 + m16;
    float bias = fb2[col];
#pragma unroll
    for (int r = 0; r < 8; ++r) {
      int orow = mt * 16 + r + 8 * half;
      if (orow < H) out[(long)orow * DD + col] = acc[nt][r] + bias;
    }
  }
}

// ---------------- launch ----------------
extern "C" void kernel_launch(void* const* d_in, const int* in_sizes, int n_in,
                              void* d_out, int out_size, void* d_ws, size_t ws_size,
                              hipStream_t stream) {
  const float* h_A = (const float*)d_in[0];
  const float* h_B = (const float*)d_in[1];
  const int*  idxA = (const int*)d_in[2];
  const int*  idxB = (const int*)d_in[3];
  // d_in[4] = num_herbs (compile-time HERBS)
  const float* aw1 = (const float*)d_in[5];
  const float* ab1 = (const float*)d_in[6];
  const float* aw2 = (const float*)d_in[7];
  const float* ab2 = (const float*)d_in[8];
  const float* fw1 = (const float*)d_in[9];
  const float* fb1 = (const float*)d_in[10];
  const float* fw2 = (const float*)d_in[11];
  const float* fb2 = (const float*)d_in[12];
  int NA = in_sizes[0] / DD;
  int NB = in_sizes[1] / DD;
  const int H = HERBS;

  char* w = (char*)d_ws;
  auto carve = [&](size_t bytes) { char* p = w; w += (bytes + 255) & ~(size_t)255; return p; };
  float* sumA   = (float*)carve(sizeof(float) * (size_t)H * DD);   // -> meanA
  float* sumB   = (float*)carve(sizeof(float) * (size_t)H * DD);   // -> meanB
  int*   cntA   = (int*)carve(sizeof(int) * H);
  int*   cntB   = (int*)carve(sizeof(int) * H);
  int*   startA = (int*)carve(sizeof(int) * H);
  int*   startB = (int*)carve(sizeof(int) * H);
  float* scoreA = (float*)carve(sizeof(float) * (size_t)NA);
  float* scoreB = (float*)carve(sizeof(float) * (size_t)NB);
  float* mid    = (float*)carve(sizeof(float) * (size_t)H * 256);
  float* fw1p   = (float*)carve(sizeof(float) * 512 * 256);        // pair-interleaved
  float* fw2p   = (float*)carve(sizeof(float) * 256 * 128);

  float* out  = (float*)d_out;
  float* HAgB = out + (size_t)H * DD;       // H_A_given_B
  float* HBgA = out + (size_t)2 * H * DD;   // H_B_given_A

  long nz = (long)H * DD;
  zero_f32_kernel<<<(int)((nz + 255) / 256), 256, 0, stream>>>(sumA, nz);
  zero_f32_kernel<<<(int)((nz + 255) / 256), 256, 0, stream>>>(sumB, nz);
  init_seg_kernel<<<(H + 255) / 256, 256, 0, stream>>>(cntA, startA, cntB, startB, H);
  pack_pairs_kernel<<<(512 * 256 + 255) / 256, 256, 0, stream>>>(fw1, fw1p, 512, 256);
  pack_pairs_kernel<<<(256 * 128 + 255) / 256, 256, 0, stream>>>(fw2, fw2p, 256, 128);

  seg_accum_kernel<<<(NA + 63) / 64, 128, 0, stream>>>(h_A, idxA, sumA, cntA, startA, NA);
  seg_accum_kernel<<<(NB + 63) / 64, 128, 0, stream>>>(h_B, idxB, sumB, cntB, startB, NB);
  seg_mean_kernel<<<H, 128, 0, stream>>>(sumA, cntA);
  seg_mean_kernel<<<H, 128, 0, stream>>>(sumB, cntB);

  int t32A = (NA + 31) / 32, t32B = (NB + 31) / 32;
  attn_score_kernel<<<(t32A + 7) / 8, 256, 0, stream>>>(h_A, idxA, sumB, aw1, ab1, aw2, ab2,
                                                        scoreA, NA, t32A);
  attn_score_kernel<<<(t32B + 7) / 8, 256, 0, stream>>>(h_B, idxB, sumA, aw1, ab1, aw2, ab2,
                                                        scoreB, NB, t32B);

  seg_attn_kernel<<<H, 128, 0, stream>>>(h_A, scoreA, startA, cntA, HAgB);
  seg_attn_kernel<<<H, 128, 0, stream>>>(h_B, scoreB, startB, cntB, HBgA);

  int mt = (H + 15) / 16;
  int jobs1 = mt * 4;
  fusion1_kernel<<<(jobs1 + 7) / 8, 256, 0, stream>>>(HAgB, HBgA, fw1p, fb1, mid, H, jobs1);
  int jobs2 = mt * 2;
  fusion2_kernel<<<(jobs2 + 7) / 8, 256, 0, stream>>>(mid, fw2p, fb2, out, H, jobs2);
}